// Decoder_54082228191476
// MI455X (gfx1250) — compile-verified
//
#include <hip/hip_runtime.h>

#define N_NODES 100000
#define N_EDGES 800000
#define DD 128
#define NEG_SLOPE 0.01f
#define N_LAYERS 2

typedef __attribute__((ext_vector_type(2))) float v2f;
typedef __attribute__((ext_vector_type(8))) float v8f;

// ---------------------------------------------------------------------------
// C = A @ B, A: [nRows x 128], B = (WT ? W^T : W), W: [128 x 128] row-major.
// One wave computes a 16-row x 128-col strip using V_WMMA_F32_16X16X4_F32.
// W is staged in LDS (64 KB; WGP has 320 KB).
//
// WMMA f32 16x16x4 VGPR layouts (ISA 7.12.2):
//   A (16x4): lanes 0-15 hold row M=lane {v0=K0, v1=K1}; lanes 16-31 {v0=K2, v1=K3}
//   B (4x16): lanes 0-15 hold col N=lane {v0=B[0][N], v1=B[1][N]}; lanes 16-31 {B[2][N], B[3][N]}
//   C/D (16x16): VGPR r: lanes 0-15 -> D[r][lane]; lanes 16-31 -> D[r+8][lane-16]
// ---------------------------------------------------------------------------
template <int WT>
__global__ __launch_bounds__(256) void gemm128_wmma(const float* __restrict__ A,
                                                    const float* __restrict__ W,
                                                    float* __restrict__ C,
                                                    int nRows) {
  __shared__ float wlds[DD * DD];
  {
    const float4* s4 = (const float4*)W;
    float4* d4 = (float4*)wlds;
    for (int i = threadIdx.x; i < DD * DD / 4; i += 256) d4[i] = s4[i];
  }
  __syncthreads();

  const int lane = threadIdx.x & 31;
  const int wave = threadIdx.x >> 5;
  const int strip = blockIdx.x * 8 + wave;
  if (strip * 16 >= nRows) return;  // no barriers after this point

  const int row0 = strip * 16;
  const int laneLo = lane & 15;
  const int hi2 = (lane >> 4) << 1;  // 0 for lanes 0-15, 2 for lanes 16-31

  const float* arow = A + (size_t)(row0 + laneLo) * DD;

  v8f acc[8] = {};

  for (int k = 0; k < DD; k += 4) {
    const int ka = k + hi2;
    v2f a;
    a.x = arow[ka];
    a.y = arow[ka + 1];
#pragma unroll
    for (int nt = 0; nt < 8; nt++) {
      const int n = nt * 16 + laneLo;
      v2f b;
      if (WT) {  // B[k][n] = W[n][k]
        b.x = wlds[n * DD + ka];
        b.y = wlds[n * DD + ka + 1];
      } else {  // B[k][n] = W[k][n]
        b.x = wlds[ka * DD + n];
        b.y = wlds[(ka + 1) * DD + n];
      }
      acc[nt] = __builtin_amdgcn_wmma_f32_16x16x4_f32(
          /*neg_a=*/false, a, /*neg_b=*/false, b,
          /*c_mod=*/(short)0, acc[nt], /*reuse_a=*/false, /*reuse_b=*/false);
    }
  }

  const int rbase = row0 + ((lane >> 4) << 3);  // +8 for hi lanes
#pragma unroll
  for (int nt = 0; nt < 8; nt++) {
#pragma unroll
    for (int r = 0; r < 8; r++) {
      C[(size_t)(rbase + r) * DD + nt * 16 + laneLo] = acc[nt][r];
    }
  }
}

// ---------------------------------------------------------------------------
// Per-node: s = z . a_src, t = z . a_dst ; also init m = -inf, denom = 0.
// One wave per node, float4 per lane, wave32 shuffle reduce.
// ---------------------------------------------------------------------------
__global__ __launch_bounds__(256) void node_attn_init(const float* __restrict__ z,
                                                      const float* __restrict__ aw,
                                                      float* __restrict__ s,
                                                      float* __restrict__ t,
                                                      float* __restrict__ m,
                                                      float* __restrict__ denom) {
  const int node = blockIdx.x * 8 + (threadIdx.x >> 5);
  const int lane = threadIdx.x & 31;
  if (node >= N_NODES) return;
  const float4 zv = ((const float4*)(z + (size_t)node * DD))[lane];
  const float4 a1 = ((const float4*)aw)[lane];        // a_src
  const float4 a2 = ((const float4*)(aw + DD))[lane]; // a_dst
  float ps = zv.x * a1.x + zv.y * a1.y + zv.z * a1.z + zv.w * a1.w;
  float pt = zv.x * a2.x + zv.y * a2.y + zv.z * a2.z + zv.w * a2.w;
#pragma unroll
  for (int off = 16; off > 0; off >>= 1) {
    ps += __shfl_down(ps, off, 32);
    pt += __shfl_down(pt, off, 32);
  }
  if (lane == 0) {
    s[node] = ps;
    t[node] = pt;
    m[node] = -__builtin_huge_valf();
    denom[node] = 0.f;
  }
}

// ---------------------------------------------------------------------------
// Per-edge: e = leakyrelu(s[src] + t[dst] + r_h . a_edge); atomicMax m[dst].
// One wave per edge (coalesced 512B row read).
// ---------------------------------------------------------------------------
__global__ __launch_bounds__(256) void edge_logits(const float* __restrict__ r_h,
                                                   const float* __restrict__ a_edge,
                                                   const float* __restrict__ s,
                                                   const float* __restrict__ t,
                                                   const int* __restrict__ src,
                                                   const int* __restrict__ dst,
                                                   float* __restrict__ e,
                                                   float* __restrict__ m) {
  const int edge = blockIdx.x * 8 + (threadIdx.x >> 5);
  const int lane = threadIdx.x & 31;
  if (edge >= N_EDGES) return;
  const float4 rv = ((const float4*)(r_h + (size_t)edge * DD))[lane];
  const float4 av = ((const float4*)a_edge)[lane];
  float p = rv.x * av.x + rv.y * av.y + rv.z * av.z + rv.w * av.w;
#pragma unroll
  for (int off = 16; off > 0; off >>= 1) p += __shfl_down(p, off, 32);
  if (lane == 0) {
    float ev = s[src[edge]] + t[dst[edge]] + p;
    ev = ev >= 0.f ? ev : NEG_SLOPE * ev;
    e[edge] = ev;
    unsafeAtomicMax(m + dst[edge], ev);  // native global_atomic_max_num_f32
  }
}

// ---------------------------------------------------------------------------
// Per-edge scalar: ex = exp(e - m[dst]); e <- ex; denom[dst] += ex.
// ---------------------------------------------------------------------------
__global__ __launch_bounds__(256) void edge_softmax(float* __restrict__ e,
                                                    const float* __restrict__ m,
                                                    float* __restrict__ denom,
                                                    const int* __restrict__ dst) {
  const int i = blockIdx.x * 256 + threadIdx.x;
  if (i >= N_EDGES) return;
  const int d = dst[i];
  const float ex = __expf(e[i] - m[d]);
  e[i] = ex;
  unsafeAtomicAdd(denom + d, ex);  // native global_atomic_add_f32
}

// ---------------------------------------------------------------------------
// Per-edge: agg[dst] += (ex/denom[dst]) * z[src].  Wave per edge, 4 cols/lane.
// ---------------------------------------------------------------------------
__global__ __launch_bounds__(256) void edge_scatter(const float* __restrict__ e,
                                                    const float* __restrict__ denom,
                                                    const float* __restrict__ z,
                                                    const int* __restrict__ src,
                                                    const int* __restrict__ dst,
                                                    float* __restrict__ agg) {
  const int edge = blockIdx.x * 8 + (threadIdx.x >> 5);
  const int lane = threadIdx.x & 31;
  if (edge >= N_EDGES) return;
  const int sI = src[edge];
  const int dI = dst[edge];
  const float alpha = e[edge] / denom[dI];
  const float4 zv = ((const float4*)(z + (size_t)sI * DD))[lane];
  float* out = agg + (size_t)dI * DD + lane * 4;
  unsafeAtomicAdd(out + 0, alpha * zv.x);
  unsafeAtomicAdd(out + 1, alpha * zv.y);
  unsafeAtomicAdd(out + 2, alpha * zv.z);
  unsafeAtomicAdd(out + 3, alpha * zv.w);
}

// ---------------------------------------------------------------------------
// h_out = relu(agg)
// ---------------------------------------------------------------------------
__global__ __launch_bounds__(256) void relu_copy(const float* __restrict__ in,
                                                 float* __restrict__ out, int n) {
  const int i = blockIdx.x * 256 + threadIdx.x;
  if (i < n) out[i] = fmaxf(in[i], 0.f);
}

// ---------------------------------------------------------------------------
extern "C" void kernel_launch(void* const* d_in, const int* in_sizes, int n_in,
                              void* d_out, int out_size, void* d_ws, size_t ws_size,
                              hipStream_t stream) {
  const float* feature = (const float*)d_in[0];
  const float* r_h     = (const float*)d_in[1];
  const float* fc_w    = (const float*)d_in[2];  // [2][128][128]
  const float* attn_w  = (const float*)d_in[3];  // [2][384]
  const float* loop_w  = (const float*)d_in[4];  // [2][128][128]
  const int*   src     = (const int*)d_in[5];
  const int*   dst     = (const int*)d_in[6];
  float* out = (float*)d_out;

  // Workspace layout (floats)
  float* z     = (float*)d_ws;                   // N*D
  float* agg   = z + (size_t)N_NODES * DD;       // N*D
  float* s     = agg + (size_t)N_NODES * DD;     // N
  float* t     = s + N_NODES;                    // N
  float* m     = t + N_NODES;                    // N
  float* denom = m + N_NODES;                    // N
  float* e     = denom + N_NODES;                // E

  const int gemmBlocks = (N_NODES / 16 + 7) / 8;   // 782
  const int nodeBlocks = (N_NODES + 7) / 8;        // wave-per-node
  const int edgeBlocks = (N_EDGES + 7) / 8;        // wave-per-edge
  const int edgeFlat   = (N_EDGES + 255) / 256;
  const int reluBlocks = (N_NODES * DD + 255) / 256;

  const float* h_in = feature;
  for (int l = 0; l < N_LAYERS; ++l) {
    const float* fc = fc_w + (size_t)l * DD * DD;
    const float* lw = loop_w + (size_t)l * DD * DD;
    const float* aw = attn_w + (size_t)l * 3 * DD;

    // z = h @ fc^T
    gemm128_wmma<1><<<gemmBlocks, 256, 0, stream>>>(h_in, fc, z, N_NODES);
    // agg = z @ lw   (self-loop term, accumulated into by edge_scatter)
    gemm128_wmma<0><<<gemmBlocks, 256, 0, stream>>>(z, lw, agg, N_NODES);
    // s, t, m=-inf, denom=0
    node_attn_init<<<nodeBlocks, 256, 0, stream>>>(z, aw, s, t, m, denom);
    // e + segment max
    edge_logits<<<edgeBlocks, 256, 0, stream>>>(r_h, aw + 2 * DD, s, t, src, dst, e, m);
    // exp + segment sum
    edge_softmax<<<edgeFlat, 256, 0, stream>>>(e, m, denom, dst);
    // weighted scatter-add of z[src]
    edge_scatter<<<edgeBlocks, 256, 0, stream>>>(e, denom, z, src, dst, agg);
    // relu -> output (d_out doubles as inter-layer h buffer)
    relu_copy<<<reluBlocks, 256, 0, stream>>>(agg, out, N_NODES * DD);

    h_in = out;
  }
}